// RwR_NN_20117626814815
// MI455X (gfx1250) — compile-verified
//
#include <hip/hip_runtime.h>

#define NNODE 2048
#define NTIME 10
#define KSTEP 5

typedef float v2f __attribute__((ext_vector_type(2)));
typedef float v8f __attribute__((ext_vector_type(8)));

// Async global->LDS copy of 16 bytes (GVS mode: saddr64 + vaddr32). ASYNCcnt.
__device__ __forceinline__ void async_b128(uint32_t lds_byte_addr,
                                           const float* base,
                                           uint32_t byte_off) {
    asm volatile("global_load_async_to_lds_b128 %0, %1, %2"
                 :: "v"(lds_byte_addr), "v"(byte_off), "s"(base)
                 : "memory");
}

// ---------------------------------------------------------------------------
// Kernel 1: per (i,t) row -> degree, G_norm, P1 = w0*Gn + (1-w0)*I, out = P1
// ---------------------------------------------------------------------------
__global__ void __launch_bounds__(256) rownorm_init_kernel(
    const float* __restrict__ x, const float* __restrict__ w,
    float* __restrict__ Gn, float* __restrict__ P1, float* __restrict__ out)
{
    __shared__ float red[256];
    const int i = blockIdx.x, t = blockIdx.y, tid = threadIdx.x;
    const float* xrow = x + (size_t)i * NTIME * NNODE + (size_t)t * NNODE;

    float s = 0.f;
#pragma unroll
    for (int r = 0; r < NNODE / 256; ++r) s += xrow[tid + r * 256];
    red[tid] = s;
    __syncthreads();
    for (int off = 128; off > 0; off >>= 1) {
        if (tid < off) red[tid] += red[tid + off];
        __syncthreads();
    }
    const float inv = 1.f / (red[0] + 1e-8f);

    float wv = w[t * KSTEP + 0];
    wv = fminf(fmaxf(wv, 0.f), 1.f);

    float* grow = Gn  + (size_t)t * NNODE * NNODE + (size_t)i * NNODE;
    float* prow = P1  + (size_t)t * NNODE * NNODE + (size_t)i * NNODE;
    float* orow = out + (size_t)i * NTIME * NNODE + (size_t)t * NNODE;
#pragma unroll
    for (int r = 0; r < NNODE / 256; ++r) {
        const int j = tid + r * 256;
        const float g = xrow[j] * inv;
        grow[j] = g;
        const float p = wv * g + ((j == i) ? (1.f - wv) : 0.f);
        prow[j] = p;
        orow[j] = p;
    }
}

// ---------------------------------------------------------------------------
// Kernel 2: batched GEMM  Pn[t] = w*(P[t] @ Gn[t]) + (1-w)*I ; out += Pn
// 128x128 C tile / block, 256 threads = 8 waves, wave = 4x2 WMMA 16x16 tiles.
// 3-stage async (GLOBAL_LOAD_ASYNC_TO_LDS_B128) pipeline, 1 barrier/slice.
// ---------------------------------------------------------------------------
__global__ void __launch_bounds__(256) rwr_gemm_kernel(
    const float* __restrict__ P, const float* __restrict__ Gn,
    float* __restrict__ Pn, float* __restrict__ out,
    const float* __restrict__ w, int kstep)
{
    constexpr int LDA = 20;   // 128x16 A tile, padded (bank-conflict free, 16B aligned)
    constexpr int LDB = 132;  // 16x128 B tile, padded
    constexpr int NBUF = 3;
    __shared__ float lA[NBUF][128 * LDA];
    __shared__ float lB[NBUF][16 * LDB];

    const int t  = blockIdx.z;
    const int bm = blockIdx.y;
    const int bn = blockIdx.x;
    const int tid   = threadIdx.x;
    const int lane  = tid & 31;
    const int wid   = tid >> 5;
    const int wm    = wid >> 2;   // 0..1  (64-row stripe)
    const int wn    = wid & 3;    // 0..3  (32-col stripe)
    const int lhalf = lane >> 4;  // 0/1   (selects K pair per ISA layout)
    const int l16   = lane & 15;  // row/col within 16x16 tile

    float wv = w[t * KSTEP + kstep];
    wv = fminf(fmaxf(wv, 0.f), 1.f);

    const size_t NN = (size_t)NNODE * NNODE;
    const float* Ag = P  + (size_t)t * NN + (size_t)(bm * 128) * NNODE; // rows [bm*128, +128)
    const float* Bg = Gn + (size_t)t * NN + (size_t)(bn * 128);         // cols [bn*128, +128)

    // LDS base byte addresses (generic-pointer low 32 bits = LDS offset)
    const uint32_t lAbase = (uint32_t)(uintptr_t)&lA[0][0];
    const uint32_t lBbase = (uint32_t)(uintptr_t)&lB[0][0];

    // Per-thread copy geometry: 2 A float4s + 2 B float4s per 16-wide K slice
    auto issue_tile = [&](int buf, int kk) {
        const uint32_t ab = lAbase + (uint32_t)buf * (128u * LDA * 4u);
        const uint32_t bb = lBbase + (uint32_t)buf * (16u * LDB * 4u);
#pragma unroll
        for (int r = 0; r < 2; ++r) {
            const int f = tid + r * 256;
            const int arow = f >> 2, akq = f & 3;
            async_b128(ab + (uint32_t)(arow * LDA + akq * 4) * 4u, Ag,
                       (uint32_t)(arow * NNODE + kk + akq * 4) * 4u);
            const int brow = f >> 5, bnq = f & 31;
            async_b128(bb + (uint32_t)(brow * LDB + bnq * 4) * 4u, Bg,
                       (uint32_t)((kk + brow) * NNODE + bnq * 4) * 4u);
        }
    };

    v8f acc[4][2] = {};

    issue_tile(0, 0);               // prologue: slice 0 -> buf 0
    int bufc = 0, bufn = 1;

    for (int kk = 0; kk < NNODE; kk += 16) {
        if (kk + 16 < NNODE) {
            issue_tile(bufn, kk + 16);                       // lookahead slice
            asm volatile("s_wait_asynccnt 0x4" ::: "memory"); // oldest 4 (= this slice) done
        } else {
            asm volatile("s_wait_asynccnt 0x0" ::: "memory");
        }
        __syncthreads();   // publish this slice's LDS data across waves

        const float* lAb = &lA[bufc][0];
        const float* lBb = &lB[bufc][0];
#pragma unroll
        for (int ks = 0; ks < 4; ++ks) {
            const int kb = ks * 4 + lhalf * 2;  // K pair for this lane half
            v2f af[4];
#pragma unroll
            for (int mi = 0; mi < 4; ++mi) {
                const int row = wm * 64 + mi * 16 + l16;
                af[mi] = *(const v2f*)&lAb[row * LDA + kb];
            }
            v2f bf[2];
#pragma unroll
            for (int ni = 0; ni < 2; ++ni) {
                const int col = wn * 32 + ni * 16 + l16;
                bf[ni].x = lBb[kb * LDB + col];
                bf[ni].y = lBb[(kb + 1) * LDB + col];
            }
#pragma unroll
            for (int mi = 0; mi < 4; ++mi)
#pragma unroll
                for (int ni = 0; ni < 2; ++ni)
                    acc[mi][ni] = __builtin_amdgcn_wmma_f32_16x16x4_f32(
                        false, af[mi], false, bf[ni], (short)0, acc[mi][ni],
                        false, false);
        }
        bufc = bufn;
        bufn = (bufn + 1 == NBUF) ? 0 : bufn + 1;
    }

    // Epilogue: v = w*C + (1-w)*I ; Pn = v ; out[i][t][j] += v
#pragma unroll
    for (int mi = 0; mi < 4; ++mi) {
#pragma unroll
        for (int ni = 0; ni < 2; ++ni) {
            const int col   = bn * 128 + wn * 32 + ni * 16 + l16;
            const int rbase = bm * 128 + wm * 64 + mi * 16 + lhalf * 8;
#pragma unroll
            for (int r = 0; r < 8; ++r) {
                const int row = rbase + r;   // C/D layout: VGPR r -> M = r + 8*(lane/16)
                const float v = wv * acc[mi][ni][r] + ((row == col) ? (1.f - wv) : 0.f);
                Pn[(size_t)t * NN + (size_t)row * NNODE + col] = v;
                out[(size_t)row * NTIME * NNODE + (size_t)t * NNODE + col] += v;
            }
        }
    }
}

// ---------------------------------------------------------------------------
// Kernel 3: in-place LayerNorm over last axis of out[i][t][:]
// ---------------------------------------------------------------------------
__global__ void __launch_bounds__(256) layernorm_kernel(
    float* __restrict__ out, const float* __restrict__ gamma,
    const float* __restrict__ beta)
{
    __shared__ float rs[256];
    __shared__ float rq[256];
    const int i = blockIdx.x, t = blockIdx.y, tid = threadIdx.x;
    float* h = out + (size_t)i * NTIME * NNODE + (size_t)t * NNODE;

    float loc[NNODE / 256];
    float s = 0.f, q = 0.f;
#pragma unroll
    for (int r = 0; r < NNODE / 256; ++r) {
        const float v = h[tid + r * 256];
        loc[r] = v; s += v; q += v * v;
    }
    rs[tid] = s; rq[tid] = q;
    __syncthreads();
    for (int off = 128; off > 0; off >>= 1) {
        if (tid < off) { rs[tid] += rs[tid + off]; rq[tid] += rq[tid + off]; }
        __syncthreads();
    }
    const float mu   = rs[0] * (1.f / NNODE);
    const float var  = rq[0] * (1.f / NNODE) - mu * mu;
    const float rstd = rsqrtf(var + 1e-5f);
#pragma unroll
    for (int r = 0; r < NNODE / 256; ++r) {
        const int j = tid + r * 256;
        h[j] = (loc[r] - mu) * rstd * gamma[j] + beta[j];
    }
}

// ---------------------------------------------------------------------------
extern "C" void kernel_launch(void* const* d_in, const int* in_sizes, int n_in,
                              void* d_out, int out_size, void* d_ws, size_t ws_size,
                              hipStream_t stream) {
    const float* x     = (const float*)d_in[0];  // [N, T, N]
    const float* wgt   = (const float*)d_in[1];  // [T, K]
    const float* gamma = (const float*)d_in[2];  // [N]
    const float* beta  = (const float*)d_in[3];  // [N]
    float* out = (float*)d_out;                  // [N, T, N]

    const size_t NN = (size_t)NNODE * NNODE;
    float* Gn = (float*)d_ws;           // [T, N, N]
    float* Pa = Gn + (size_t)NTIME * NN;
    float* Pb = Pa + (size_t)NTIME * NN;

    rownorm_init_kernel<<<dim3(NNODE, NTIME), 256, 0, stream>>>(x, wgt, Gn, Pa, out);

    float* Pc = Pa;
    float* Pn = Pb;
    for (int k = 1; k < KSTEP; ++k) {
        rwr_gemm_kernel<<<dim3(NNODE / 128, NNODE / 128, NTIME), 256, 0, stream>>>(
            Pc, Gn, Pn, out, wgt, k);
        float* tmp = Pc; Pc = Pn; Pn = tmp;
    }

    layernorm_kernel<<<dim3(NNODE, NTIME), 256, 0, stream>>>(out, gamma, beta);
}